// GT32dim_18708877541404
// MI455X (gfx1250) — compile-verified
//
#include <hip/hip_runtime.h>
#include <math.h>
#include <stdint.h>

typedef __attribute__((ext_vector_type(2))) float v2f;
typedef __attribute__((ext_vector_type(8))) float v8f;

// ---------------------------------------------------------------------------
// Fused 2-matrix GEMM:  OA = H@WA+bA, OB = H@WB+bB    (both [nrows,32])
// f32 WMMA 16x16x4, one wave per 16-row tile, grid-stride over tiles.
// B fragments are tile-invariant -> fully unrolled K lets the compiler hoist
// all 64 B VGPRs out of the tile loop (fits: 32 accum + 64 B + ~32 misc).
// Layout (ISA 7.12.2):
//   A: lane m=lane%16 holds row m; k = kc + 2*(lane/16) + {0,1}  (float2 load)
//   B: lane n=lane%16 holds col n (+16 for tile 1), same k striping
//   C/D: VGPR g holds row g + 8*(lane/16), col lane%16 (+16)
// ---------------------------------------------------------------------------
template <int K>
__global__ __launch_bounds__(256)
void gemm2_wmma(const float* __restrict__ H, int ldH,
                const float* __restrict__ WA, const float* __restrict__ bA,
                const float* __restrict__ WB, const float* __restrict__ bB,
                float* __restrict__ OA, float* __restrict__ OB,
                int nrows)
{
    const int lane = threadIdx.x & 31;
    const int wv   = threadIdx.x >> 5;
    const int wpb  = blockDim.x >> 5;
    const int m    = lane & 15;
    const int kg   = lane >> 4;
    const int ntiles = (nrows + 15) >> 4;

    for (int t = blockIdx.x * wpb + wv; t < ntiles; t += gridDim.x * wpb) {
        const int row0 = t << 4;
        const int r    = row0 + m;
        const int rc   = r < (nrows - 1) ? r : (nrows - 1);   // clamped, always valid
        const float mask = (r < nrows) ? 1.f : 0.f;
        const float* __restrict__ Hrow = H + (size_t)rc * ldH;

        v8f accA0 = {0,0,0,0,0,0,0,0}, accA1 = {0,0,0,0,0,0,0,0};
        v8f accB0 = {0,0,0,0,0,0,0,0}, accB1 = {0,0,0,0,0,0,0,0};

        #pragma unroll
        for (int kc = 0; kc < K; kc += 4) {
            const int k0 = kc + (kg << 1);
            float2 h2 = *(const float2*)(Hrow + k0);
            v2f a; a.x = h2.x * mask; a.y = h2.y * mask;

            const float* A0 = WA + (size_t)k0 * 32;
            const float* B0 = WB + (size_t)k0 * 32;
            v2f bA0, bA1, bB0, bB1;
            bA0.x = A0[m];      bA0.y = A0[32 + m];
            bA1.x = A0[16 + m]; bA1.y = A0[48 + m];
            bB0.x = B0[m];      bB0.y = B0[32 + m];
            bB1.x = B0[16 + m]; bB1.y = B0[48 + m];

            accA0 = __builtin_amdgcn_wmma_f32_16x16x4_f32(false, a, false, bA0, (short)0, accA0, false, false);
            accA1 = __builtin_amdgcn_wmma_f32_16x16x4_f32(false, a, false, bA1, (short)0, accA1, false, false);
            accB0 = __builtin_amdgcn_wmma_f32_16x16x4_f32(false, a, false, bB0, (short)0, accB0, false, false);
            accB1 = __builtin_amdgcn_wmma_f32_16x16x4_f32(false, a, false, bB1, (short)0, accB1, false, false);
        }

        const float ba0 = bA[m], ba1 = bA[16 + m];
        const float bb0 = bB[m], bb1 = bB[16 + m];
        #pragma unroll
        for (int g = 0; g < 8; ++g) {
            const int rr = row0 + g + (kg << 3);
            if (rr < nrows) {
                const size_t o = (size_t)rr * 32;
                OA[o + m] = accA0[g] + ba0;  OA[o + 16 + m] = accA1[g] + ba1;
                OB[o + m] = accB0[g] + bb0;  OB[o + 16 + m] = accB1[g] + bb1;
            }
        }
    }
}

// ---------------------------------------------------------------------------
// Single GEMM: OUT[nrows,32] = H[nrows,K] @ W[K,32] + bias  (W1, K=128)
// unroll capped at 4 so live B fragments stay bounded (no spills).
// ---------------------------------------------------------------------------
template <int K>
__global__ __launch_bounds__(256)
void gemm_wmma_n32(const float* __restrict__ H, int ldH,
                   const float* __restrict__ W,
                   const float* __restrict__ bias,
                   float* __restrict__ OUT, int ldOut,
                   int nrows)
{
    const int lane = threadIdx.x & 31;
    const int wv   = threadIdx.x >> 5;
    const int wpb  = blockDim.x >> 5;
    const int m    = lane & 15;
    const int kg   = lane >> 4;
    const int ntiles = (nrows + 15) >> 4;

    for (int t = blockIdx.x * wpb + wv; t < ntiles; t += gridDim.x * wpb) {
        const int row0 = t << 4;
        const int r    = row0 + m;
        const int rc   = r < (nrows - 1) ? r : (nrows - 1);
        const float mask = (r < nrows) ? 1.f : 0.f;
        const float* __restrict__ Hrow = H + (size_t)rc * ldH;

        v8f acc0 = {0,0,0,0,0,0,0,0};
        v8f acc1 = {0,0,0,0,0,0,0,0};

        #pragma unroll 4
        for (int kc = 0; kc < K; kc += 4) {
            const int k0 = kc + (kg << 1);
            float2 h2 = *(const float2*)(Hrow + k0);
            v2f a; a.x = h2.x * mask; a.y = h2.y * mask;
            const float* W0 = W + (size_t)k0 * 32;
            v2f b0, b1;
            b0.x = W0[m];        b0.y = W0[32 + m];
            b1.x = W0[16 + m];   b1.y = W0[48 + m];
            acc0 = __builtin_amdgcn_wmma_f32_16x16x4_f32(false, a, false, b0, (short)0, acc0, false, false);
            acc1 = __builtin_amdgcn_wmma_f32_16x16x4_f32(false, a, false, b1, (short)0, acc1, false, false);
        }
        const float bc0 = bias[m], bc1 = bias[16 + m];
        #pragma unroll
        for (int g = 0; g < 8; ++g) {
            const int rr = row0 + g + (kg << 3);
            if (rr < nrows) {
                OUT[(size_t)rr * ldOut + m]      = acc0[g] + bc0;
                OUT[(size_t)rr * ldOut + 16 + m] = acc1[g] + bc1;
            }
        }
    }
}

// ---- monotone float <-> uint mapping for atomicMax on f32 ------------------
__device__ __forceinline__ unsigned int ford(float f) {
    unsigned int u = __float_as_uint(f);
    return (u & 0x80000000u) ? ~u : (u | 0x80000000u);
}
__device__ __forceinline__ float funord(unsigned int u) {
    return __uint_as_float((u & 0x80000000u) ? (u & 0x7FFFFFFFu) : ~u);
}

// alpha[e] = scale * dot(q[dst], k[src]); atomicMax per-dst running max
__global__ void edge_alpha_max(const long long* __restrict__ src,
                               const long long* __restrict__ dst,
                               const float* __restrict__ q,
                               const float* __restrict__ k,
                               float* __restrict__ alpha,
                               unsigned int* __restrict__ amaxU,
                               int E, float scale)
{
    int e = blockIdx.x * blockDim.x + threadIdx.x;
    if (e >= E) return;
    const int s = (int)src[e], d = (int)dst[e];
    const float4* qd = (const float4*)(q + (size_t)d * 32);
    const float4* ks = (const float4*)(k + (size_t)s * 32);
    float acc = 0.f;
    #pragma unroll
    for (int i = 0; i < 8; ++i) {
        float4 a = qd[i], b = ks[i];
        acc += a.x*b.x + a.y*b.y + a.z*b.z + a.w*b.w;
    }
    acc *= scale;
    alpha[e] = acc;
    atomicMax(&amaxU[d], ford(acc));
}

// ex = exp(alpha - amax[dst]); denom[dst]+=ex; agg[dst] += ex * v[src]
__global__ void edge_exp_scatter(const long long* __restrict__ src,
                                 const long long* __restrict__ dst,
                                 const float* __restrict__ vbuf,
                                 const float* __restrict__ alpha,
                                 const unsigned int* __restrict__ amaxU,
                                 float* __restrict__ denom,
                                 float* __restrict__ agg,
                                 int E)
{
    int e = blockIdx.x * blockDim.x + threadIdx.x;
    if (e >= E) return;
    const int s = (int)src[e], d = (int)dst[e];
    const float ex = __expf(alpha[e] - funord(amaxU[d]));
    atomicAdd(&denom[d], ex);
    const float4* vs = (const float4*)(vbuf + (size_t)s * 32);
    float* ad = agg + (size_t)d * 32;
    #pragma unroll
    for (int i = 0; i < 8; ++i) {
        float4 vv = vs[i];
        atomicAdd(ad + 4*i + 0, ex * vv.x);
        atomicAdd(ad + 4*i + 1, ex * vv.y);
        atomicAdd(ad + 4*i + 2, ex * vv.z);
        atomicAdd(ad + 4*i + 3, ex * vv.w);
    }
}

// h = tanh(agg/denom + skip); write into feats column block for this layer
__global__ void node_finalize(const float* __restrict__ agg,
                              const float* __restrict__ denom,
                              const float* __restrict__ skip,
                              float* __restrict__ featsCol, // feats + 32*l
                              int N)
{
    int idx = blockIdx.x * blockDim.x + threadIdx.x;
    if (idx >= N * 32) return;
    int n = idx >> 5, j = idx & 31;
    float dn  = denom[n];
    float inv = (dn > 0.f) ? (1.f / dn) : 0.f;   // empty segment -> 0 (matches ref)
    float h = tanhf(agg[idx] * inv + skip[idx]);
    featsCol[(size_t)n * 128 + j] = h;
}

// scatter per-node [N,32] into per-graph sums + counts
__global__ void pool_scatter(const float* __restrict__ hn,
                             const long long* __restrict__ batch,
                             float* __restrict__ sums, float* __restrict__ cnt,
                             int N)
{
    int idx = blockIdx.x * blockDim.x + threadIdx.x;
    if (idx >= N * 32) return;
    int n = idx >> 5, j = idx & 31;
    int g = (int)batch[n];
    atomicAdd(&sums[g * 32 + j], hn[idx]);
    if (j == 0) atomicAdd(&cnt[g], 1.0f);
}

// per-graph head: mean-pool -> relu(W2) -> log_softmax(W3). One block per graph.
__global__ void head_kernel(const float* __restrict__ sums,
                            const float* __restrict__ cnt,
                            const float* __restrict__ W2, const float* __restrict__ b2,
                            const float* __restrict__ W3, const float* __restrict__ b3,
                            float* __restrict__ out, int C)
{
    __shared__ float pooled[32];
    __shared__ float h2[128];
    __shared__ float logits[16];
    const int g = blockIdx.x, t = threadIdx.x;
    if (t < 32) {
        float c = cnt[g]; c = c > 1.f ? c : 1.f;
        pooled[t] = sums[g * 32 + t] / c;
    }
    __syncthreads();
    {   // t = 0..127
        float acc = b2[t];
        #pragma unroll 4
        for (int j = 0; j < 32; ++j) acc += pooled[j] * W2[j * 128 + t];
        h2[t] = acc > 0.f ? acc : 0.f;
    }
    __syncthreads();
    if (t < C) {
        float a = b3[t];
        for (int j = 0; j < 128; ++j) a += h2[j] * W3[j * C + t];
        logits[t] = a;
    }
    __syncthreads();
    if (t == 0) {
        float mx = logits[0];
        for (int c = 1; c < C; ++c) mx = fmaxf(mx, logits[c]);
        float se = 0.f;
        for (int c = 0; c < C; ++c) se += __expf(logits[c] - mx);
        float lse = mx + logf(se);
        for (int c = 0; c < C; ++c) out[g * C + c] = logits[c] - lse;
    }
}

// ---------------------------------------------------------------------------
extern "C" void kernel_launch(void* const* d_in, const int* in_sizes, int n_in,
                              void* d_out, int out_size, void* d_ws, size_t ws_size,
                              hipStream_t stream)
{
    const float*     x     = (const float*)d_in[0];
    const long long* eidx  = (const long long*)d_in[1];   // int64 [2, E]
    const long long* batch = (const long long*)d_in[2];   // int64 [N]
    const float* Wq = (const float*)d_in[3];  const float* bq = (const float*)d_in[4];
    const float* Wk = (const float*)d_in[5];  const float* bk = (const float*)d_in[6];
    const float* Wv = (const float*)d_in[7];  const float* bv = (const float*)d_in[8];
    const float* Wsk= (const float*)d_in[9];  const float* bsk= (const float*)d_in[10];
    const float* W1 = (const float*)d_in[11]; const float* b1 = (const float*)d_in[12];
    const float* W2 = (const float*)d_in[13]; const float* b2 = (const float*)d_in[14];
    const float* W3 = (const float*)d_in[15]; const float* b3 = (const float*)d_in[16];

    const int N = in_sizes[2];
    const int E = in_sizes[1] / 2;
    const int C = in_sizes[15] / 128;          // W3 is [128, C]
    const int G = out_size / C;
    const int L = 4;
    const long long* src = eidx;
    const long long* dst = eidx + E;

    // ---- workspace carve-up (256B aligned) ----
    size_t off = 0;
    auto carve = [&](size_t bytes) -> void* {
        void* p = (char*)d_ws + off;
        off += (bytes + 255) & ~(size_t)255;
        return p;
    };
    float*        feats = (float*)carve((size_t)N * 128 * 4); // concat of 4 layer outputs
    float*        qb    = (float*)carve((size_t)N * 32 * 4);
    float*        kb    = (float*)carve((size_t)N * 32 * 4);
    float*        vb    = (float*)carve((size_t)N * 32 * 4);
    float*        sb    = (float*)carve((size_t)N * 32 * 4);
    float*        agg   = (float*)carve((size_t)N * 32 * 4);
    float*        alpha = (float*)carve((size_t)E * 4);
    unsigned int* amaxU = (unsigned int*)carve((size_t)N * 4);
    float*        denom = (float*)carve((size_t)N * 4);
    float*        hn    = (float*)carve((size_t)N * 32 * 4);
    float*        sums  = (float*)carve((size_t)G * 32 * 4);
    float*        cnt   = (float*)carve((size_t)G * 4);
    (void)ws_size; (void)n_in;

    const float scale = 0.17677669529663688f;  // 1/sqrt(32)
    const int ntiles  = (N + 15) / 16;
    const int gemmBlk = 256;                   // 8 waves/block
    const int gemmGrd = (ntiles + 7) / 8;
    const int eGrd    = (E + 255) / 256;
    const int nGrd    = (N * 32 + 255) / 256;

    for (int l = 0; l < L; ++l) {
        const float* Hsrc; const float* Hdst; int ldH;
        if (l == 0) { Hsrc = x; Hdst = x + 32; ldH = 64; }         // x_src | x_dst halves
        else        { Hsrc = Hdst = feats + 32 * (l - 1); ldH = 128; }

        // q,s from Hdst ; k,v from Hsrc  (two fused 2-matrix GEMMs)
        gemm2_wmma<32><<<gemmGrd, gemmBlk, 0, stream>>>(
            Hdst, ldH, Wq + l * 1024, bq + l * 32, Wsk + l * 1024, bsk + l * 32,
            qb, sb, N);
        gemm2_wmma<32><<<gemmGrd, gemmBlk, 0, stream>>>(
            Hsrc, ldH, Wk + l * 1024, bk + l * 32, Wv + l * 1024, bv + l * 32,
            kb, vb, N);

        hipMemsetAsync(amaxU, 0, (size_t)N * 4, stream);           // 0 == floor below all finite
        hipMemsetAsync(denom, 0, (size_t)N * 4, stream);
        hipMemsetAsync(agg,   0, (size_t)N * 32 * 4, stream);

        edge_alpha_max<<<eGrd, 256, 0, stream>>>(src, dst, qb, kb, alpha, amaxU, E, scale);
        edge_exp_scatter<<<eGrd, 256, 0, stream>>>(src, dst, vb, alpha, amaxU, denom, agg, E);
        node_finalize<<<nGrd, 256, 0, stream>>>(agg, denom, sb, feats + 32 * l, N);
    }

    // cs @ W1 + b1  -> hn [N, 32]   (K = 128)
    gemm_wmma_n32<128><<<gemmGrd, gemmBlk, 0, stream>>>(feats, 128, W1, b1, hn, 32, N);

    hipMemsetAsync(sums, 0, (size_t)G * 32 * 4, stream);
    hipMemsetAsync(cnt,  0, (size_t)G * 4, stream);
    pool_scatter<<<nGrd, 256, 0, stream>>>(hn, batch, sums, cnt, N);

    head_kernel<<<G, 128, 0, stream>>>(sums, cnt, W2, b2, W3, b3, (float*)d_out, C);
}